// Encoder_72335839200000
// MI455X (gfx1250) — compile-verified
//
#include <hip/hip_runtime.h>
#include <hip/hip_bf16.h>
#include <math.h>

typedef __attribute__((ext_vector_type(2))) float v2f;
typedef __attribute__((ext_vector_type(8))) float v8f;

#define GAT_HEADS 4
#define GAT_NEG_SLOPE 0.2f

// ---------------------------------------------------------------------------
// Full-precision WMMA GEMM:  C[M,N] = A[M,K] @ B[K,N] (+ bias[N] if non-null)
// One wave32 per 16x32 output slab (two adjacent 16x16 tiles), K stepped by 4
// via V_WMMA_F32_16X16X4_F32. Each A fragment (contiguous float2 per lane,
// ISA 7.12.2: lane m=lane&15 row, lane group g=lane>>4 selects K pair) feeds
// two WMMA issues -> register-level A reuse, 2x WMMA per loop body.
// C/D layout: VGPR r -> row r (lanes 0-15) or r+8 (lanes 16-31), col=lane&15.
// ---------------------------------------------------------------------------
__global__ void wmma_gemm_f32(const float* __restrict__ A,
                              const float* __restrict__ B,
                              const float* __restrict__ bias,
                              float* __restrict__ C,
                              int M, int N, int K) {
  const int lane  = threadIdx.x & 31;
  const int wave  = blockIdx.x * (blockDim.x >> 5) + (threadIdx.x >> 5);
  const int npair = N >> 5;                 // pairs of 16-wide N tiles
  const int mt    = wave / npair;
  const int np    = wave - mt * npair;
  if (mt * 16 >= M) return;

  const int m = lane & 15;   // A row within tile / C column lane
  const int g = lane >> 4;   // lane group: selects K pair and C row half

  const float* __restrict__ Arow = A + (size_t)(mt * 16 + m) * K;
  const float* __restrict__ B0   = B + np * 32 + m;
  const float* __restrict__ B1   = B0 + 16;

  v8f acc0 = {}, acc1 = {};
  for (int k0 = 0; k0 < K; k0 += 4) {
    const int ka = k0 + 2 * g;
    v2f a, b0, b1;
    a.x  = Arow[ka];
    a.y  = Arow[ka + 1];
    b0.x = B0[(size_t)ka * N];
    b0.y = B0[(size_t)(ka + 1) * N];
    b1.x = B1[(size_t)ka * N];
    b1.y = B1[(size_t)(ka + 1) * N];
    acc0 = __builtin_amdgcn_wmma_f32_16x16x4_f32(false, a, false, b0,
                                                 (short)0, acc0, false, false);
    acc1 = __builtin_amdgcn_wmma_f32_16x16x4_f32(false, a, false, b1,
                                                 (short)0, acc1, false, false);
  }

  const int col0 = np * 32 + m;
  const float bv0 = bias ? bias[col0] : 0.0f;
  const float bv1 = bias ? bias[col0 + 16] : 0.0f;
  float* __restrict__ Crow = C + (size_t)(mt * 16 + 8 * g) * N + col0;
#pragma unroll
  for (int r = 0; r < 8; ++r) {
    Crow[(size_t)r * N]      = acc0[r] + bv0;
    Crow[(size_t)r * N + 16] = acc1[r] + bv1;
  }
}

// ---------------------------------------------------------------------------
// Attention logits: al_s[n,h] = <xh[n,h,:], a_src[h,:]>, same for a_dst.
// One thread per (node, head), float4-vectorized inner product.
// ---------------------------------------------------------------------------
__global__ void attn_logits(const float* __restrict__ xh,
                            const float* __restrict__ a_src,
                            const float* __restrict__ a_dst,
                            float* __restrict__ al_s,
                            float* __restrict__ al_d,
                            int Nn, int C) {
  const int t = blockIdx.x * blockDim.x + threadIdx.x;
  const int node = t >> 2, h = t & 3;
  if (node >= Nn) return;
  const float4* __restrict__ v4 =
      (const float4*)(xh + ((size_t)node * GAT_HEADS + h) * C);
  const float4* __restrict__ as4 = (const float4*)(a_src + h * C);
  const float4* __restrict__ ad4 = (const float4*)(a_dst + h * C);
  float s = 0.0f, d = 0.0f;
  const int C4 = C >> 2;
  for (int c = 0; c < C4; ++c) {
    const float4 x = v4[c], a = as4[c], b = ad4[c];
    s = fmaf(x.x, a.x, fmaf(x.y, a.y, fmaf(x.z, a.z, fmaf(x.w, a.w, s))));
    d = fmaf(x.x, b.x, fmaf(x.y, b.y, fmaf(x.z, b.z, fmaf(x.w, b.w, d))));
  }
  al_s[t] = s;
  al_d[t] = d;
}

// monotonic float <-> uint key so unsigned atomicMax implements float max
__device__ __forceinline__ unsigned f2key(float f) {
  unsigned u = __float_as_uint(f);
  return (u & 0x80000000u) ? ~u : (u | 0x80000000u);
}
__device__ __forceinline__ float key2f(unsigned k) {
  return (k & 0x80000000u) ? __uint_as_float(k & 0x7FFFFFFFu)
                           : __uint_as_float(~k);
}
__device__ __forceinline__ float leaky(float a) {
  return (a > 0.0f) ? a : a * GAT_NEG_SLOPE;
}

// pass 1: alpha = leaky_relu(al_s[src]+al_d[dst]); segment max over dst.
// One thread per edge; all 4 heads via float4.
__global__ void edge_alpha_max(const float4* __restrict__ al_s,
                               const float4* __restrict__ al_d,
                               const int* __restrict__ ei,
                               float4* __restrict__ alpha,
                               unsigned* __restrict__ amaxk,
                               int E, int Etot) {
  const int e = blockIdx.x * blockDim.x + threadIdx.x;
  if (e >= Etot) return;
  int s, d;
  if (e < E) { s = ei[e]; d = ei[E + e]; } else { s = e - E; d = s; }
  const float4 as = al_s[s];
  const float4 ad = al_d[d];
  float4 a;
  a.x = leaky(as.x + ad.x);
  a.y = leaky(as.y + ad.y);
  a.z = leaky(as.z + ad.z);
  a.w = leaky(as.w + ad.w);
  alpha[e] = a;
  unsigned* mk = amaxk + (size_t)d * GAT_HEADS;
  atomicMax(&mk[0], f2key(a.x));
  atomicMax(&mk[1], f2key(a.y));
  atomicMax(&mk[2], f2key(a.z));
  atomicMax(&mk[3], f2key(a.w));
}

// pass 2: ex = exp(alpha - max[dst]); segment sum of ex over dst.
__global__ void edge_expsum(const int* __restrict__ ei,
                            float4* __restrict__ alpha,
                            const uint4* __restrict__ amaxk,
                            float* __restrict__ denom,
                            int E, int Etot) {
  const int e = blockIdx.x * blockDim.x + threadIdx.x;
  if (e >= Etot) return;
  int d;
  if (e < E) { d = ei[E + e]; } else { d = e - E; }
  const uint4 mk = amaxk[d];
  float4 a = alpha[e];
  a.x = __expf(a.x - key2f(mk.x));
  a.y = __expf(a.y - key2f(mk.y));
  a.z = __expf(a.z - key2f(mk.z));
  a.w = __expf(a.w - key2f(mk.w));
  alpha[e] = a;
  float* dn = denom + (size_t)d * GAT_HEADS;
  atomicAdd(&dn[0], a.x);
  atomicAdd(&dn[1], a.y);
  atomicAdd(&dn[2], a.z);
  atomicAdd(&dn[3], a.w);
}

// pass 3: acc[dst] += xh[src] * coef. One wave32 per edge, float4 gathers.
__global__ void edge_scatter(const float4* __restrict__ xh4,
                             const float4* __restrict__ ex4,
                             const float* __restrict__ denom,
                             const int* __restrict__ ei,
                             float* __restrict__ acc,
                             int E, int Etot, int HC) {
  const int e = blockIdx.x * (blockDim.x >> 5) + (threadIdx.x >> 5);
  if (e >= Etot) return;
  const int lane = threadIdx.x & 31;
  int s, d;
  if (e < E) { s = ei[e]; d = ei[E + e]; } else { s = e - E; d = s; }
  const int Q  = HC >> 2;   // float4s per node row (64 or 32)
  const int C4 = Q >> 2;    // float4s per head (16 or 8)
  const float4* __restrict__ sp = xh4 + (size_t)s * Q;
  float* __restrict__ dp = acc + (size_t)d * HC;
  __builtin_prefetch((const void*)(sp + lane), 0, 1);  // global_prefetch_b8
  const float4 exv = ex4[e];
  const float* __restrict__ dn = denom + (size_t)d * GAT_HEADS;
  float coef[GAT_HEADS];
  coef[0] = exv.x / (dn[0] + 1e-16f);
  coef[1] = exv.y / (dn[1] + 1e-16f);
  coef[2] = exv.z / (dn[2] + 1e-16f);
  coef[3] = exv.w / (dn[3] + 1e-16f);
  for (int q = lane; q < Q; q += 32) {
    const float4 v = sp[q];
    const float c = coef[q / C4];
    atomicAdd(&dp[q * 4 + 0], v.x * c);
    atomicAdd(&dp[q * 4 + 1], v.y * c);
    atomicAdd(&dp[q * 4 + 2], v.z * c);
    atomicAdd(&dp[q * 4 + 3], v.w * c);
  }
}

// mean over heads + bias (+ optional ReLU), float4-vectorized
__global__ void combine_heads(const float4* __restrict__ acc4,
                              const float4* __restrict__ bias4,
                              float4* __restrict__ out4,
                              int Nn, int C4, int do_relu) {
  const int t = blockIdx.x * blockDim.x + threadIdx.x;
  if (t >= Nn * C4) return;
  const int node = t / C4, c = t - node * C4;
  const float4* __restrict__ p = acc4 + (size_t)node * GAT_HEADS * C4 + c;
  float4 s = {0.0f, 0.0f, 0.0f, 0.0f};
#pragma unroll
  for (int h = 0; h < GAT_HEADS; ++h) {
    const float4 v = p[(size_t)h * C4];
    s.x += v.x; s.y += v.y; s.z += v.z; s.w += v.w;
  }
  const float4 b = bias4[c];
  float4 o;
  o.x = s.x * (1.0f / GAT_HEADS) + b.x;
  o.y = s.y * (1.0f / GAT_HEADS) + b.y;
  o.z = s.z * (1.0f / GAT_HEADS) + b.z;
  o.w = s.w * (1.0f / GAT_HEADS) + b.w;
  if (do_relu) {
    o.x = fmaxf(o.x, 0.0f); o.y = fmaxf(o.y, 0.0f);
    o.z = fmaxf(o.z, 0.0f); o.w = fmaxf(o.w, 0.0f);
  }
  out4[t] = o;
}

// ---------------------------------------------------------------------------
extern "C" void kernel_launch(void* const* d_in, const int* in_sizes, int n_in,
                              void* d_out, int out_size, void* d_ws, size_t ws_size,
                              hipStream_t stream) {
  const float* x   = (const float*)d_in[0];
  const int*   ei  = (const int*)  d_in[1];
  const float* W1  = (const float*)d_in[2];
  const float* as1 = (const float*)d_in[3];
  const float* ad1 = (const float*)d_in[4];
  const float* b1  = (const float*)d_in[5];
  const float* W2  = (const float*)d_in[6];
  const float* as2 = (const float*)d_in[7];
  const float* ad2 = (const float*)d_in[8];
  const float* b2  = (const float*)d_in[9];
  const float* Wmu = (const float*)d_in[10];
  const float* bmu = (const float*)d_in[11];
  const float* Wlv = (const float*)d_in[12];
  const float* blv = (const float*)d_in[13];

  const int IN_DIM = 128, HID = 64, HID2 = 32, LATENT = 32;
  const int Nn   = in_sizes[0] / IN_DIM;  // 50000
  const int E    = in_sizes[1] / 2;       // 800000
  const int Etot = E + Nn;                // + self loops

  // ---- workspace layout (manual, with cross-layer reuse) ----
  char* ws = (char*)d_ws;
  size_t off = 0;
  auto wsalloc = [&](size_t bytes) -> void* {
    void* p = ws + off;
    off += (bytes + 255) & ~(size_t)255;
    return p;
  };
  float*    XH   = (float*)   wsalloc((size_t)Nn * GAT_HEADS * HID * 4); // xh1 / xh2
  float*    ACC  = (float*)   wsalloc((size_t)Nn * GAT_HEADS * HID * 4); // acc1 / acc2
  float*    H1   = (float*)   wsalloc((size_t)Nn * HID  * 4);
  float*    H2   = (float*)   wsalloc((size_t)Nn * HID2 * 4);
  float*    ALPH = (float*)   wsalloc((size_t)Etot * GAT_HEADS * 4);
  float*    ALS  = (float*)   wsalloc((size_t)Nn * GAT_HEADS * 4);
  float*    ALD  = (float*)   wsalloc((size_t)Nn * GAT_HEADS * 4);
  unsigned* AMAX = (unsigned*)wsalloc((size_t)Nn * GAT_HEADS * 4);
  float*    DEN  = (float*)   wsalloc((size_t)Nn * GAT_HEADS * 4);
  (void)ws_size; (void)n_in;

  const int TPB = 256;
  const int edgeBlocks     = (Etot + TPB - 1) / TPB;  // per-edge kernels
  const int edgeWaveBlocks = (Etot + 7) / 8;          // 8 edge-waves per block

  // ================= Layer 1 =================
  {
    const int M = Nn, Nd = GAT_HEADS * HID, K = IN_DIM;
    const int waves = (M / 16) * (Nd / 32);
    wmma_gemm_f32<<<(waves + 7) / 8, TPB, 0, stream>>>(x, W1, nullptr, XH, M, Nd, K);
    attn_logits<<<(Nn * 4 + TPB - 1) / TPB, TPB, 0, stream>>>(XH, as1, ad1, ALS, ALD, Nn, HID);
    hipMemsetAsync(AMAX, 0, (size_t)Nn * GAT_HEADS * 4, stream);
    hipMemsetAsync(DEN,  0, (size_t)Nn * GAT_HEADS * 4, stream);
    hipMemsetAsync(ACC,  0, (size_t)Nn * GAT_HEADS * HID * 4, stream);
    edge_alpha_max<<<edgeBlocks, TPB, 0, stream>>>((const float4*)ALS, (const float4*)ALD,
                                                   ei, (float4*)ALPH, AMAX, E, Etot);
    edge_expsum<<<edgeBlocks, TPB, 0, stream>>>(ei, (float4*)ALPH, (const uint4*)AMAX,
                                                DEN, E, Etot);
    edge_scatter<<<edgeWaveBlocks, TPB, 0, stream>>>((const float4*)XH, (const float4*)ALPH,
                                                     DEN, ei, ACC, E, Etot, GAT_HEADS * HID);
    combine_heads<<<(Nn * (HID / 4) + TPB - 1) / TPB, TPB, 0, stream>>>(
        (const float4*)ACC, (const float4*)b1, (float4*)H1, Nn, HID / 4, 1);
  }

  // ================= Layer 2 =================
  {
    const int M = Nn, Nd = GAT_HEADS * HID2, K = HID;
    const int waves = (M / 16) * (Nd / 32);
    wmma_gemm_f32<<<(waves + 7) / 8, TPB, 0, stream>>>(H1, W2, nullptr, XH, M, Nd, K);
    attn_logits<<<(Nn * 4 + TPB - 1) / TPB, TPB, 0, stream>>>(XH, as2, ad2, ALS, ALD, Nn, HID2);
    hipMemsetAsync(AMAX, 0, (size_t)Nn * GAT_HEADS * 4, stream);
    hipMemsetAsync(DEN,  0, (size_t)Nn * GAT_HEADS * 4, stream);
    hipMemsetAsync(ACC,  0, (size_t)Nn * GAT_HEADS * HID2 * 4, stream);
    edge_alpha_max<<<edgeBlocks, TPB, 0, stream>>>((const float4*)ALS, (const float4*)ALD,
                                                   ei, (float4*)ALPH, AMAX, E, Etot);
    edge_expsum<<<edgeBlocks, TPB, 0, stream>>>(ei, (float4*)ALPH, (const uint4*)AMAX,
                                                DEN, E, Etot);
    edge_scatter<<<edgeWaveBlocks, TPB, 0, stream>>>((const float4*)XH, (const float4*)ALPH,
                                                     DEN, ei, ACC, E, Etot, GAT_HEADS * HID2);
    combine_heads<<<(Nn * (HID2 / 4) + TPB - 1) / TPB, TPB, 0, stream>>>(
        (const float4*)ACC, (const float4*)b2, (float4*)H2, Nn, HID2 / 4, 1);
  }

  // ================= mu / logvar heads =================
  {
    float* mu = (float*)d_out;
    float* lv = mu + (size_t)Nn * LATENT;
    const int waves = (Nn / 16) * (LATENT / 32);
    wmma_gemm_f32<<<(waves + 7) / 8, TPB, 0, stream>>>(H2, Wmu, bmu, mu, Nn, LATENT, HID2);
    wmma_gemm_f32<<<(waves + 7) / 8, TPB, 0, stream>>>(H2, Wlv, blv, lv, Nn, LATENT, HID2);
  }
  (void)out_size;
}